// GCN_48636209660237
// MI455X (gfx1250) — compile-verified
//
#include <hip/hip_runtime.h>

typedef float v2f __attribute__((ext_vector_type(2)));
typedef float v8f __attribute__((ext_vector_type(8)));

#define N_NODES 30000
#define D_DEG   32
#define H_DIM   128
#define K_LAB   10
#define K_AUG   144        // 128 (n1_h) + 10 (label) + 2 (x) + Ppos + Pneg + bias + pad
#define N_STEPS (K_AUG / 4)   // 36 wmma k-steps
#define AS_STRIDE 148      // 16B-aligned rows; 20*row mod 64 injective -> conflict-free

// ---------------------------------------------------------------------------
// Prep (1 block, 256 threads):
// Phase A (threads 0..127): build dense augmented weights Bdense[144][128]:
//   rows   0..127 : W2
//   rows 128..137 : W1
//   rows 138..139 : W0
//   row  140      : u[h'] = sum_h relu(Wt4[h]) * Wt3[h][h']  (rank-2 of s@Wt3)
//   row  141      : v[h'] = sum_h min(Wt4[h],0) * Wt3[h][h']
//   row  142      : b0 + b1 + b2
//   row  143      : 0
// Phase B (all 256 threads): swizzle into per-lane WMMA fragment order:
//   Bsw[(tile*32 + lane)*72 + t*2 + {0,1}] = Bdense[4t + 2*(lane>>4) + {0,1}][tile*16 + (lane&15)]
// so each lane's 36 fragments are 72 contiguous floats -> b128 loads.
// ---------------------------------------------------------------------------
__global__ __launch_bounds__(256) void gcn_prep(
    const float* __restrict__ W0, const float* __restrict__ b0,
    const float* __restrict__ W1, const float* __restrict__ b1,
    const float* __restrict__ W2, const float* __restrict__ b2,
    const float* __restrict__ Wt3, const float* __restrict__ Wt4,
    float* __restrict__ Bdense, float* __restrict__ Bsw) {
  const int tid = threadIdx.x;
  if (tid < H_DIM) {
    const int hp = tid;                  // output column
    for (int k = 0; k < H_DIM; ++k)
      Bdense[k * H_DIM + hp] = W2[k * H_DIM + hp];
    for (int k = 0; k < K_LAB; ++k)
      Bdense[(H_DIM + k) * H_DIM + hp] = W1[k * H_DIM + hp];
    Bdense[138 * H_DIM + hp] = W0[hp];
    Bdense[139 * H_DIM + hp] = W0[H_DIM + hp];
    float u = 0.f, v = 0.f;
    for (int h = 0; h < H_DIM; ++h) {
      float t  = Wt4[h];
      float w3 = Wt3[h * H_DIM + hp];
      u += fmaxf(t, 0.f) * w3;
      v += fminf(t, 0.f) * w3;
    }
    Bdense[140 * H_DIM + hp] = u;
    Bdense[141 * H_DIM + hp] = v;
    Bdense[142 * H_DIM + hp] = b0[hp] + b1[hp] + b2[hp];
    Bdense[143 * H_DIM + hp] = 0.f;
  }
  __syncthreads();

  const int tile  = tid >> 5;            // 0..7 column tile
  const int lane  = tid & 31;
  const int n     = tile * 16 + (lane & 15);
  const int khalf = 2 * (lane >> 4);
  float* dst = Bsw + (tile * 32 + lane) * (2 * N_STEPS);
  for (int t = 0; t < N_STEPS; ++t) {
    const int kk = 4 * t + khalf;
    dst[2 * t + 0] = Bdense[kk * H_DIM + n];
    dst[2 * t + 1] = Bdense[(kk + 1) * H_DIM + n];
  }
}

// ---------------------------------------------------------------------------
// Main: one block = 16 nodes, 256 threads (8 waves).
//  phase 0: stage 512 edges into LDS
//  phase 1: gather n1_h with float4 loads (feature table is L2-resident),
//           build augmented A row in LDS
//  phase 2: 8 waves x one 16x16 tile, K=144 via v_wmma_f32_16x16x4_f32,
//           B fragments streamed as b128 from the pre-swizzled layout,
//           fused ReLU + store.
// ---------------------------------------------------------------------------
__global__ __launch_bounds__(256) void gcn_main(
    const float* __restrict__ feature, const float* __restrict__ x,
    const float* __restrict__ label,   const float* __restrict__ w,
    const float* __restrict__ e_type,  const int* __restrict__ src,
    const float* __restrict__ Bsw,     float* __restrict__ out) {

  __shared__ int   s_src[16 * D_DEG];
  __shared__ float s_et0[16 * D_DEG];
  __shared__ float s_A[16 * AS_STRIDE];

  const int tid = threadIdx.x;
  const int nb  = blockIdx.x * 16;       // first node of this tile

  // ---- phase 0: stage edge data (512 edges) into LDS ----
  for (int i = tid; i < 16 * D_DEG; i += 256) {
    const int n = nb + (i >> 5);
    const int d = i & (D_DEG - 1);
    s_src[i] = src[n * D_DEG + d];
    s_et0[i] = e_type[(n * D_DEG + d) * 2 + 0];
  }
  __syncthreads();

  // ---- phase 1: gather-weighted sum; thread owns 8 contiguous columns ----
  const int row  = tid >> 4;             // 0..15 node-in-tile
  const int c    = tid & 15;
  const int node = nb + row;

  float acc[8];
#pragma unroll
  for (int j = 0; j < 8; ++j) acc[j] = 0.f;
  for (int d = 0; d < D_DEG; ++d) {
    const int   sidx = s_src[row * D_DEG + d];
    const float e0   = s_et0[row * D_DEG + d];
    const float4* frow4 = (const float4*)(feature + (long)sidx * H_DIM) + c * 2;
    const float4 f0 = frow4[0];
    const float4 f1 = frow4[1];
    acc[0] += f0.x * e0; acc[1] += f0.y * e0;
    acc[2] += f0.z * e0; acc[3] += f0.w * e0;
    acc[4] += f1.x * e0; acc[5] += f1.y * e0;
    acc[6] += f1.z * e0; acc[7] += f1.w * e0;
  }
#pragma unroll
  for (int j = 0; j < 8; ++j) s_A[row * AS_STRIDE + c * 8 + j] = acc[j];

  // ---- phase 1b: augmented columns 128..143 ----
  if (c < K_LAB) {
    s_A[row * AS_STRIDE + H_DIM + c] = label[node * K_LAB + c];
  } else if (c == 10) {
    s_A[row * AS_STRIDE + 138] = x[node * 2 + 0];
  } else if (c == 11) {
    s_A[row * AS_STRIDE + 139] = x[node * 2 + 1];
  } else if (c == 12) {
    s_A[row * AS_STRIDE + 142] = 1.0f;   // bias slot
  } else if (c == 13) {
    s_A[row * AS_STRIDE + 143] = 0.0f;   // pad slot
  } else if (c == 14) {                  // Ppos = sum_d max(w*et0, 0)
    float p = 0.f;
    for (int d = 0; d < D_DEG; ++d) {
      float nw = w[node * D_DEG + d] * s_et0[row * D_DEG + d];
      p += fmaxf(nw, 0.f);
    }
    s_A[row * AS_STRIDE + 140] = p;
  } else {                               // c == 15: Pneg = sum_d min(w*et0, 0)
    float q = 0.f;
    for (int d = 0; d < D_DEG; ++d) {
      float nw = w[node * D_DEG + d] * s_et0[row * D_DEG + d];
      q += fminf(nw, 0.f);
    }
    s_A[row * AS_STRIDE + 141] = q;
  }
  __syncthreads();

  // ---- phase 2: D = relu(A_aug(16x144) @ B_aug(144x128)), WMMA f32 ----
  const int wave = tid >> 5;             // 0..7 -> output column tile
  const int lane = tid & 31;
  const int m    = lane & 15;            // A row / D column for this lane
  const int hi   = lane >> 4;            // half-wave: K pair select / M half
  const int ncol = wave * 16 + (lane & 15);

  const float4* bfr =
      (const float4*)(Bsw + (wave * 32 + lane) * (2 * N_STEPS));

  v8f cacc = {0.f, 0.f, 0.f, 0.f, 0.f, 0.f, 0.f, 0.f};
#pragma unroll
  for (int t2 = 0; t2 < N_STEPS / 2; ++t2) {
    const float4 bv = bfr[t2];           // fragments for steps 2*t2, 2*t2+1
    {
      const int kk = 4 * (2 * t2) + 2 * hi;
      v2f a, b;
      a.x = s_A[m * AS_STRIDE + kk];
      a.y = s_A[m * AS_STRIDE + kk + 1];
      b.x = bv.x; b.y = bv.y;
      cacc = __builtin_amdgcn_wmma_f32_16x16x4_f32(
          false, a, false, b, (short)0, cacc, false, false);
    }
    {
      const int kk = 4 * (2 * t2 + 1) + 2 * hi;
      v2f a, b;
      a.x = s_A[m * AS_STRIDE + kk];
      a.y = s_A[m * AS_STRIDE + kk + 1];
      b.x = bv.z; b.y = bv.w;
      cacc = __builtin_amdgcn_wmma_f32_16x16x4_f32(
          false, a, false, b, (short)0, cacc, false, false);
    }
  }

  // fused ReLU + store (D layout: VGPR r -> M = r + 8*hi, N = lane&15)
#pragma unroll
  for (int r = 0; r < 8; ++r) {
    const int mrow = r + 8 * hi;
    const float vr = cacc[r];
    out[(long)(nb + mrow) * H_DIM + ncol] = vr > 0.f ? vr : 0.f;
  }
}

extern "C" void kernel_launch(void* const* d_in, const int* in_sizes, int n_in,
                              void* d_out, int out_size, void* d_ws, size_t ws_size,
                              hipStream_t stream) {
  const float* feature = (const float*)d_in[0];
  const float* x       = (const float*)d_in[1];
  const float* label   = (const float*)d_in[2];
  const float* w       = (const float*)d_in[3];
  const float* e_type  = (const float*)d_in[4];
  const int*   src     = (const int*)  d_in[5];
  const float* W0      = (const float*)d_in[6];
  const float* b0      = (const float*)d_in[7];
  const float* W1      = (const float*)d_in[8];
  const float* b1      = (const float*)d_in[9];
  const float* W2      = (const float*)d_in[10];
  const float* b2      = (const float*)d_in[11];
  const float* Wt3     = (const float*)d_in[12];
  const float* Wt4     = (const float*)d_in[13];

  float* Bdense = (float*)d_ws;                          // 144*128*4 = 73728 B
  float* Bsw    = (float*)((char*)d_ws + K_AUG * H_DIM * sizeof(float));  // 73728 B

  gcn_prep<<<1, 256, 0, stream>>>(W0, b0, W1, b1, W2, b2, Wt3, Wt4,
                                  Bdense, Bsw);
  gcn_main<<<N_NODES / 16, 256, 0, stream>>>(feature, x, label, w, e_type, src,
                                             Bsw, (float*)d_out);
}